// PatchFFTLoss_7988639170720
// MI455X (gfx1250) — compile-verified
//
#include <hip/hip_runtime.h>

// ---------------------------------------------------------------------------
// PatchFFT loss for (16,3,512,512) fp32, 8x8 patches, via f32 WMMA DFT GEMMs.
// fftshift dropped (permutation-invariant under the final mean).
// ---------------------------------------------------------------------------

typedef float v2f __attribute__((ext_vector_type(2)));
typedef float v8f __attribute__((ext_vector_type(8)));

#define INV_SQRT8 0.35355339059327373f
#define N_TOTAL_INV (1.0f / 12582912.0f)   // 16*3*512*512 loss elements

__constant__ float COS8[8] = { 1.f,  0.70710678118654752f,  0.f, -0.70710678118654752f,
                              -1.f, -0.70710678118654752f,  0.f,  0.70710678118654752f};
__constant__ float SIN8[8] = { 0.f,  0.70710678118654752f,  1.f,  0.70710678118654752f,
                               0.f, -0.70710678118654752f, -1.f, -0.70710678118654752f};

__device__ __forceinline__ float dftR(int j, int k) { return  COS8[(j * k) & 7] * INV_SQRT8; }
__device__ __forceinline__ float dftI(int j, int k) { return -SIN8[(j * k) & 7] * INV_SQRT8; }

// sign(x) * log1p(|x|)
__device__ __forceinline__ float nlog(float x) {
  return __builtin_copysignf(__logf(1.0f + __builtin_fabsf(x)), x);
}

#define WMMA_F32(A, B, C) \
  __builtin_amdgcn_wmma_f32_16x16x4_f32(false, (A), false, (B), (short)0, (C), false, false)

__global__ __launch_bounds__(256) void patchfft_loss_kernel(
    const float* __restrict__ in, const float* __restrict__ tg,
    float* __restrict__ out, int npairs) {
  // Per-wave transpose staging: 8 waves x (8 rows x 33 floats, padded vs banks)
  __shared__ float trIn[8][264];
  __shared__ float trTg[8][264];

  const int lane = threadIdx.x & 31;
  const int wib  = threadIdx.x >> 5;   // wave in block
  const int lm   = lane & 15;
  const int lh   = lane >> 4;          // 0 or 1

  // ---- lane-resident constant operands --------------------------------
  // Stage-1 A operand: [Fr; Fi] (16x8). A f32 16x4 layout:
  //   VGPR v, lane l -> (M = l%16, K = (l/16)*2 + v); chunk t covers K=4t..4t+3.
  v2f a1[2];
#pragma unroll
  for (int t = 0; t < 2; ++t)
#pragma unroll
    for (int v = 0; v < 2; ++v) {
      int k = t * 4 + lh * 2 + v;
      a1[t][v] = (lm < 8) ? dftR(lm, k) : dftI(lm - 8, k);
    }

  // Stage-2 B operand: [[Fr^T, Fi^T], [-Fi^T, Fr^T]] (16x16). B f32 4x16 layout
  // assumed mirror of A: VGPR v, lane l -> (K = (l/16)*2 + v, N = l%16).
  v2f b2[4];
#pragma unroll
  for (int t = 0; t < 4; ++t)
#pragma unroll
    for (int v = 0; v < 2; ++v) {
      int K = t * 4 + lh * 2 + v;
      int N = lm;
      float val;
      if (K < 8) val = (N < 8) ?  dftR(N, K)     : dftI(N - 8, K);
      else       val = (N < 8) ? -dftI(N, K - 8) : dftR(N - 8, K - 8);
      b2[t][v] = val;
    }

  // LDS gather indices converting stage-1 D (C/D layout) into stage-2 A layout.
  // D element (M,N): M<8 -> row M lane N ; M>=8 -> row M-8 lane 16+N.
  // A2(m,K): m<8 patch1, m>=8 patch2; K<8 real half, K>=8 imag half.
  int a2idx[4][2];
#pragma unroll
  for (int t = 0; t < 4; ++t)
#pragma unroll
    for (int v = 0; v < 2; ++v) {
      int k = t * 4 + lh * 2 + v;
      a2idx[t][v] = (lm & 7) * 33 + (lm >> 3) * 8 + (k >> 3) * 16 + (k & 7);
    }

  const int totalWaves = (int)((gridDim.x * blockDim.x) >> 5);
  const int waveId     = (int)((blockIdx.x * blockDim.x + threadIdx.x) >> 5);

  float acc = 0.0f;
  const v8f zero8 = {};

  for (int q = waveId; q < npairs; q += totalWaves) {
    // pair q -> two horizontally adjacent 8x8 patches (16 contiguous cols)
    const int plane = q >> 11;           // 2048 pairs per (b,c) plane
    const int rem   = q & 2047;
    const int pr    = rem >> 5;          // patch row (0..63)
    const int pc    = rem & 31;          // pair column (0..31)
    const size_t base = (size_t)plane * 262144 + (size_t)pr * 4096 + (size_t)pc * 16;
    const float* pin = in + base;
    const float* ptg = tg + base;

    // Prefetch next tile while we compute this one.
    int qn = q + totalWaves;
    if (qn < npairs) {
      const size_t basen = (size_t)(qn >> 11) * 262144 +
                           (size_t)((qn & 2047) >> 5) * 4096 +
                           (size_t)(qn & 31) * 16;
      __builtin_prefetch(in + basen, 0, 0);
      __builtin_prefetch(tg + basen, 0, 0);
    }

    // ---- stage-1 B operands: row k of [X1|X2] = 16 contiguous floats ----
    v2f bin[2], btg[2];
#pragma unroll
    for (int t = 0; t < 2; ++t)
#pragma unroll
      for (int v = 0; v < 2; ++v) {
        int k = t * 4 + lh * 2 + v;
        bin[t][v] = pin[(size_t)k * 512 + lm];
        btg[t][v] = ptg[(size_t)k * 512 + lm];
      }

    // ---- stage 1: Y = F * X  (D1 = [Yr1 Yr2; Yi1 Yi2]) ------------------
    v8f yin = zero8, ytg = zero8;
    yin = WMMA_F32(a1[0], bin[0], yin);
    yin = WMMA_F32(a1[1], bin[1], yin);
    ytg = WMMA_F32(a1[0], btg[0], ytg);
    ytg = WMMA_F32(a1[1], btg[1], ytg);

    // ---- transpose D1 -> A2 via LDS (same-wave, in-order DS) ------------
#pragma unroll
    for (int r = 0; r < 8; ++r) {
      trIn[wib][r * 33 + lane] = yin[r];
      trTg[wib][r * 33 + lane] = ytg[r];
    }
    __builtin_amdgcn_wave_barrier();
    asm volatile("s_wait_dscnt 0" ::: "memory");

    v2f a2in[4], a2tg[4];
#pragma unroll
    for (int t = 0; t < 4; ++t)
#pragma unroll
      for (int v = 0; v < 2; ++v) {
        a2in[t][v] = trIn[wib][a2idx[t][v]];
        a2tg[t][v] = trTg[wib][a2idx[t][v]];
      }
    __builtin_amdgcn_wave_barrier();

    // ---- stage 2: Z = Y * F^T  (D2 = [Zr1 Zi1; Zr2 Zi2]) ----------------
    v8f zin = zero8, ztg = zero8;
#pragma unroll
    for (int t = 0; t < 4; ++t) {
      zin = WMMA_F32(a2in[t], b2[t], zin);
      ztg = WMMA_F32(a2tg[t], b2[t], ztg);
    }

    // ---- elementwise sign*log1p + squared diff (layout-agnostic) --------
#pragma unroll
    for (int r = 0; r < 8; ++r) {
      float d = nlog(zin[r]) - nlog(ztg[r]);
      acc = __builtin_fmaf(d, d, acc);
    }
  }

  // ---- reduction: wave32 shfl tree, then one atomic per wave ------------
  acc *= N_TOTAL_INV;
#pragma unroll
  for (int off = 16; off > 0; off >>= 1)
    acc += __shfl_xor(acc, off, 32);
  if (lane == 0) atomicAdd(out, acc);
}

extern "C" void kernel_launch(void* const* d_in, const int* in_sizes, int n_in,
                              void* d_out, int out_size, void* d_ws, size_t ws_size,
                              hipStream_t stream) {
  (void)in_sizes; (void)n_in; (void)out_size; (void)d_ws; (void)ws_size;
  const float* in = (const float*)d_in[0];
  const float* tg = (const float*)d_in[1];
  float* out = (float*)d_out;

  // Harness poisons d_out; zero the accumulator deterministically each call.
  hipMemsetAsync(out, 0, sizeof(float), stream);

  const int npairs = 98304;  // 196,608 patches / 2
  patchfft_loss_kernel<<<1024, 256, 0, stream>>>(in, tg, out, npairs);
}